// Aggregator_hyp_29300266893689
// MI455X (gfx1250) — compile-verified
//
#include <hip/hip_runtime.h>

#define WAVES_PER_BLOCK 8
#define THREADS_PER_BLOCK (WAVES_PER_BLOCK * 32)
#define DFEAT 128

#if defined(__AMDGCN__) && defined(__gfx1250__) && \
    __has_builtin(__builtin_amdgcn_global_load_async_to_lds_b32)
#define USE_ASYNC_LDS 1
typedef __attribute__((address_space(1))) int  as1_int;
typedef __attribute__((address_space(3))) int  as3_int;
#else
#define USE_ASYNC_LDS 0
#endif

// Scalar (SMEM-pipe) binary search: `key` must be wave-uniform; the uniform
// address lets the backend select s_load_b32 instead of vector loads.
__device__ __forceinline__ int lower_bound_uniform(const int* __restrict__ a, int n, int key) {
  int lo = 0, hi = n;
  while (lo < hi) {
    int mid = (lo + hi) >> 1;
    int v = a[mid];                 // uniform address -> s_load_b32
    if (v < key) lo = mid + 1; else hi = mid;
  }
  return lo;
}

// One wave (32 lanes) per output row. Lane l accumulates features [4l, 4l+4).
// Edge (val, col) pairs are staged into per-wave LDS slabs with the gfx1250
// async-to-LDS path; x-row gathers are 512B fully-coalesced b128 loads.
__global__ __launch_bounds__(THREADS_PER_BLOCK)
void coo_spmm_rowwave_kernel(const float* __restrict__ x,
                             const float* __restrict__ vals,
                             const int*   __restrict__ rows,
                             const int*   __restrict__ cols,
                             float*       __restrict__ out,
                             int n_nodes, int n_edges) {
  __shared__ float s_vals[WAVES_PER_BLOCK][32];
  __shared__ int   s_cols[WAVES_PER_BLOCK][32];

  const int lane = threadIdx.x & 31;
  const int wave = threadIdx.x >> 5;
  const int r = blockIdx.x * WAVES_PER_BLOCK + wave;
  if (r >= n_nodes) return;   // whole-wave exit only

  // Force wave-uniformity so the binary search runs on the scalar pipe.
  const int ru = __builtin_amdgcn_readfirstlane(r);
  const int start = lower_bound_uniform(rows, n_edges, ru);
  const int end   = lower_bound_uniform(rows, n_edges, ru + 1);

  float4 acc = make_float4(0.f, 0.f, 0.f, 0.f);
  const float* xlane = x + (lane << 2);        // this lane's 16B feature slice

  float* s_vals_w = s_vals[wave];
  int*   s_cols_w = s_cols[wave];

  for (int e0 = start; e0 < end; e0 += 32) {
    const int n = (end - e0 < 32) ? (end - e0) : 32;

#if USE_ASYNC_LDS
    if (lane < n) {
      __builtin_amdgcn_global_load_async_to_lds_b32(
          (as1_int*)(vals + e0 + lane), (as3_int*)(s_vals_w + lane), 0, 0);
      __builtin_amdgcn_global_load_async_to_lds_b32(
          (as1_int*)(cols + e0 + lane), (as3_int*)(s_cols_w + lane), 0, 0);
    }
#if __has_builtin(__builtin_amdgcn_s_wait_asynccnt)
    __builtin_amdgcn_s_wait_asynccnt(0);
#else
    asm volatile("s_wait_asynccnt 0" ::: "memory");
#endif
#else
    if (lane < n) {
      s_vals_w[lane] = vals[e0 + lane];
      s_cols_w[lane] = cols[e0 + lane];
    }
    __builtin_amdgcn_wave_barrier();  // same-wave LDS ops are in-order
#endif

    int k = 0;
    // 2x unrolled: two b128 gathers in flight before the FMA chain waits.
    for (; k + 1 < n; k += 2) {
      const float v0 = s_vals_w[k];
      const float v1 = s_vals_w[k + 1];
      const unsigned c0 = (unsigned)s_cols_w[k];
      const unsigned c1 = (unsigned)s_cols_w[k + 1];
      const float4 x0 = *reinterpret_cast<const float4*>(xlane + (c0 << 7));
      const float4 x1 = *reinterpret_cast<const float4*>(xlane + (c1 << 7));
      acc.x = fmaf(v0, x0.x, acc.x);
      acc.y = fmaf(v0, x0.y, acc.y);
      acc.z = fmaf(v0, x0.z, acc.z);
      acc.w = fmaf(v0, x0.w, acc.w);
      acc.x = fmaf(v1, x1.x, acc.x);
      acc.y = fmaf(v1, x1.y, acc.y);
      acc.z = fmaf(v1, x1.z, acc.z);
      acc.w = fmaf(v1, x1.w, acc.w);
    }
    if (k < n) {
      const float v = s_vals_w[k];
      const unsigned c = (unsigned)s_cols_w[k];
      const float4 xv = *reinterpret_cast<const float4*>(xlane + (c << 7));
      acc.x = fmaf(v, xv.x, acc.x);
      acc.y = fmaf(v, xv.y, acc.y);
      acc.z = fmaf(v, xv.z, acc.z);
      acc.w = fmaf(v, xv.w, acc.w);
    }
  }

  // Rows with no edges correctly store zeros (segment_sum semantics).
  *reinterpret_cast<float4*>(out + ((size_t)(unsigned)r << 7) + (lane << 2)) = acc;
}

extern "C" void kernel_launch(void* const* d_in, const int* in_sizes, int n_in,
                              void* d_out, int out_size, void* d_ws, size_t ws_size,
                              hipStream_t stream) {
  const float* x    = (const float*)d_in[0];
  const float* vals = (const float*)d_in[1];
  const int*   rows = (const int*)d_in[2];
  const int*   cols = (const int*)d_in[3];
  float* out = (float*)d_out;

  const int n_nodes = in_sizes[0] / DFEAT;
  const int n_edges = in_sizes[1];

  const int blocks = (n_nodes + WAVES_PER_BLOCK - 1) / WAVES_PER_BLOCK;
  hipLaunchKernelGGL(coo_spmm_rowwave_kernel, dim3(blocks), dim3(THREADS_PER_BLOCK),
                     0, stream, x, vals, rows, cols, out, n_nodes, n_edges);
}